// DecoderNoAR_49675591745627
// MI455X (gfx1250) — compile-verified
//
#include <hip/hip_runtime.h>
#include <hip/hip_bf16.h>
#include <math.h>
#include <stdint.h>

// ---------------------------------------------------------------------------
// MI455X (gfx1250) implementation. Bandwidth-bound workload (~620 MB moved,
// ~22 GFLOP) -> keep fp32, use V_WMMA_F32_16X16X4_F32 for all dense GEMMs,
// and GLOBAL_LOAD_ASYNC_TO_LDS_B128 (ASYNCcnt) for contiguous LDS staging.
// ---------------------------------------------------------------------------

typedef float v2f __attribute__((ext_vector_type(2)));
typedef float v8f __attribute__((ext_vector_type(8)));

#define FCH 64
#define NMAXQ 16
#define LN_EPS 1e-5f
#define VN_EPS 1e-6f
#define MOLS_PER_WAVE 5   // 5 molecules x 3 v-dims = 15 cols + 1 zero pad = 16

__device__ __forceinline__ v8f wmma_f32_4(v2f a, v2f b, v8f c) {
  // D = A(16x4, f32) * B(4x16, f32) + C(16x16, f32)
  return __builtin_amdgcn_wmma_f32_16x16x4_f32(
      /*neg_a=*/false, a, /*neg_b=*/false, b,
      /*c_mod=*/(short)0, c, /*reuse_a=*/false, /*reuse_b=*/false);
}

// ---- async global->LDS copy (gfx1250 GLOBAL_LOAD_ASYNC_TO_LDS_B128) -------
// Builtin prototype (from compiler diagnostic):
//   void __builtin_amdgcn_global_load_async_to_lds_b128(
//       int __attribute__((vector_size(16))) * AS1 gsrc,
//       int __attribute__((vector_size(16))) * AS3 ldst,
//       int imm_offset, int imm_cpol);
#if __has_builtin(__builtin_amdgcn_global_load_async_to_lds_b128)
#define HAVE_ASYNC_LDS 1
typedef int v4i_gcc __attribute__((vector_size(16)));
typedef __attribute__((address_space(1))) v4i_gcc* g_v4i_p;
typedef __attribute__((address_space(3))) v4i_gcc* l_v4i_p;
__device__ __forceinline__ void async_b128(const void* g, void* l) {
  // generic->AS1 is numerically identity; generic LDS pointer's low 32 bits
  // are the LDS byte offset (aperture in the high bits).
  __builtin_amdgcn_global_load_async_to_lds_b128(
      (g_v4i_p)(uintptr_t)g, (l_v4i_p)(uint32_t)(uintptr_t)l,
      /*offset=*/0, /*cpol=*/0);
}
__device__ __forceinline__ void async_wait0() {
#if __has_builtin(__builtin_amdgcn_s_wait_asynccnt)
  __builtin_amdgcn_s_wait_asynccnt(0);
#else
  asm volatile("s_wait_asynccnt 0" ::: "memory");
#endif
}
#else
#define HAVE_ASYNC_LDS 0
#endif

// Copy `n4` float4s from global `src` to LDS `dst` with one wave (32 lanes).
__device__ __forceinline__ void wave_copy_f4(float* dst, const float* src,
                                             int n4, int lane) {
  const float4* s4 = (const float4*)src;
  float4* d4 = (float4*)dst;
#if HAVE_ASYNC_LDS
  for (int i = lane; i < n4; i += 32) async_b128(s4 + i, d4 + i);
  async_wait0();
#else
  for (int i = lane; i < n4; i += 32) d4[i] = s4[i];
#endif
}

// ===========================================================================
// Kernel 1: per-molecule VN-MLP (W_ff1/W_ff2/W_ffd via WMMA) + vn_relu +
// residual + vn_layernorm + 16x64 attention. One wave (32 threads) per block,
// 5 molecules per wave.
// ===========================================================================
__global__ __launch_bounds__(32)
void vn_attn_kernel(const float* __restrict__ Z_V,
                    const float* __restrict__ x_fine,
                    const float* __restrict__ W_ff1,
                    const float* __restrict__ W_ff2,
                    const float* __restrict__ W_ffd,
                    const float* __restrict__ g1v,
                    const float* __restrict__ b1v,
                    const float* __restrict__ ipw,
                    const float* __restrict__ ipb,
                    const float* __restrict__ outw,
                    const float* __restrict__ outb,
                    const int*   __restrict__ frag_idx,
                    const int*   __restrict__ counts,
                    float*       __restrict__ attn_out,
                    int N)
{
  __shared__ __attribute__((aligned(16))) float Wbuf[128 * 64]; // reused per stage
  __shared__ __attribute__((aligned(16))) float Xs[64 * 16];    // X then final Z
  __shared__ __attribute__((aligned(16))) float H1[128 * 16];   // H1, then D matrix
  __shared__ __attribute__((aligned(16))) float H2[64 * 16];    // H2, then residual
  __shared__ float norms[MOLS_PER_WAVE * 64];
  __shared__ float Kb[64 * 3], Vb[64 * 3];
  __shared__ float sIpw[27], sIpb[9], sOw[9], sOb[3], sG[64], sB[64];
  __shared__ float sMu[MOLS_PER_WAVE], sRs[MOLS_PER_WAVE];

  const int lane = threadIdx.x;      // wave32, blockDim == 32
  const int base = blockIdx.x * MOLS_PER_WAVE;
  const int hh = lane >> 4;          // lane half: selects K pair in WMMA tiles
  const int nn = lane & 15;

  // stage tiny params
  if (lane < 27) sIpw[lane] = ipw[lane];
  if (lane < 9)  { sIpb[lane] = ipb[lane]; sOw[lane] = outw[lane]; }
  if (lane < 3)  sOb[lane] = outb[lane];
  sG[lane] = g1v[lane]; sG[lane + 32] = g1v[lane + 32];
  sB[lane] = b1v[lane]; sB[lane + 32] = b1v[lane + 32];

  // ---- X: (64 ch x 16 cols), col = 3*m + v, coalesced load of Z_V ----
  for (int e = lane; e < MOLS_PER_WAVE * 192; e += 32) {
    int m = e / 192, r = e % 192;
    int n = base + m; if (n > N - 1) n = N - 1;     // clamp tail
    float v = Z_V[(long)n * 192 + r];
    int c = r / 3, vv = r % 3;
    Xs[c * 16 + m * 3 + vv] = v;
  }
  for (int c = lane; c < 64; c += 32) Xs[c * 16 + 15] = 0.f; // pad col
  __syncthreads();

  // ---- stage 1: H1(128x16) = W_ff1(128x64) @ X(64x16) ----
  wave_copy_f4(Wbuf, W_ff1, 128 * 64 / 4, lane);
  __syncthreads();
  for (int rt = 0; rt < 8; ++rt) {
    v8f acc = {0.f, 0.f, 0.f, 0.f, 0.f, 0.f, 0.f, 0.f};
    for (int kk = 0; kk < 16; ++kk) {
      int k0 = kk * 4 + 2 * hh;
      int m = rt * 16 + nn;
      v2f a; a.x = Wbuf[m * 64 + k0];     a.y = Wbuf[m * 64 + k0 + 1];
      v2f b; b.x = Xs[k0 * 16 + nn];      b.y = Xs[(k0 + 1) * 16 + nn];
      acc = wmma_f32_4(a, b, acc);
    }
#pragma unroll
    for (int p = 0; p < 8; ++p)
      H1[(rt * 16 + p + 8 * hh) * 16 + nn] = acc[p];
  }
  __syncthreads();

  // ---- stage 2: H2(64x16) = W_ff2(64x128) @ H1(128x16) ----
  wave_copy_f4(Wbuf, W_ff2, 64 * 128 / 4, lane);
  __syncthreads();
  for (int rt = 0; rt < 4; ++rt) {
    v8f acc = {0.f, 0.f, 0.f, 0.f, 0.f, 0.f, 0.f, 0.f};
    for (int kk = 0; kk < 32; ++kk) {
      int k0 = kk * 4 + 2 * hh;
      int m = rt * 16 + nn;
      v2f a; a.x = Wbuf[m * 128 + k0];    a.y = Wbuf[m * 128 + k0 + 1];
      v2f b; b.x = H1[k0 * 16 + nn];      b.y = H1[(k0 + 1) * 16 + nn];
      acc = wmma_f32_4(a, b, acc);
    }
#pragma unroll
    for (int p = 0; p < 8; ++p)
      H2[(rt * 16 + p + 8 * hh) * 16 + nn] = acc[p];
  }
  __syncthreads();

  // ---- stage 3: D(64x16) = W_ffd(64x64) @ H2(64x16); D aliases H1 ----
  wave_copy_f4(Wbuf, W_ffd, 64 * 64 / 4, lane);
  __syncthreads();
  float* Dm = H1;
  for (int rt = 0; rt < 4; ++rt) {
    v8f acc = {0.f, 0.f, 0.f, 0.f, 0.f, 0.f, 0.f, 0.f};
    for (int kk = 0; kk < 16; ++kk) {
      int k0 = kk * 4 + 2 * hh;
      int m = rt * 16 + nn;
      v2f a; a.x = Wbuf[m * 64 + k0];     a.y = Wbuf[m * 64 + k0 + 1];
      v2f b; b.x = H2[k0 * 16 + nn];      b.y = H2[(k0 + 1) * 16 + nn];
      acc = wmma_f32_4(a, b, acc);
    }
#pragma unroll
    for (int p = 0; p < 8; ++p)
      Dm[(rt * 16 + p + 8 * hh) * 16 + nn] = acc[p];
  }
  __syncthreads();

  // ---- vn_relu + residual + per-channel norms (v-dim = 3) ----
  for (int ci = 0; ci < 2; ++ci) {
    int c = lane + 32 * ci;
    for (int m = 0; m < MOLS_PER_WAVE; ++m) {
      int o = c * 16 + m * 3;
      float x0 = H2[o], x1 = H2[o + 1], x2 = H2[o + 2];
      float d0 = Dm[o], d1 = Dm[o + 1], d2 = Dm[o + 2];
      float dot = x0 * d0 + x1 * d1 + x2 * d2;
      float dsq = d0 * d0 + d1 * d1 + d2 * d2;
      float tf = dot / (dsq + VN_EPS);
      float y0 = (dot >= 0.f) ? x0 : x0 - tf * d0;
      float y1 = (dot >= 0.f) ? x1 : x1 - tf * d1;
      float y2 = (dot >= 0.f) ? x2 : x2 - tf * d2;
      float r0 = y0 + Xs[o], r1 = y1 + Xs[o + 1], r2 = y2 + Xs[o + 2];
      H2[o] = r0; H2[o + 1] = r1; H2[o + 2] = r2;
      norms[m * 64 + c] = sqrtf(r0 * r0 + r1 * r1 + r2 * r2);
    }
  }
  __syncthreads();
  if (lane < MOLS_PER_WAVE) {
    float s = 0.f, s2 = 0.f;
    for (int c = 0; c < 64; ++c) {
      float nv = norms[lane * 64 + c];
      s += nv; s2 += nv * nv;
    }
    float mean = s * (1.f / 64.f);
    float var = s2 * (1.f / 64.f) - mean * mean;
    sMu[lane] = mean;
    sRs[lane] = rsqrtf(var + LN_EPS);
  }
  __syncthreads();
  // ---- vn_layernorm scale: Z written back into Xs ----
  for (int ci = 0; ci < 2; ++ci) {
    int c = lane + 32 * ci;
    for (int m = 0; m < MOLS_PER_WAVE; ++m) {
      int o = c * 16 + m * 3;
      float nv = norms[m * 64 + c];
      float ln = (nv - sMu[m]) * sRs[m] * sG[c] + sB[c];
      float f = ln / (nv + VN_EPS);
      Xs[o]     = H2[o] * f;
      Xs[o + 1] = H2[o + 1] * f;
      Xs[o + 2] = H2[o + 2] * f;
    }
  }
  __syncthreads();

  // ---- attention per molecule: K/V projection then 16 q-lanes x 64 keys ----
  const float rs3 = 0.5773502691896258f; // 1/sqrt(3)
  for (int m = 0; m < MOLS_PER_WAVE; ++m) {
    for (int ci = 0; ci < 2; ++ci) {
      int c = lane + 32 * ci;
      int o = c * 16 + m * 3;
      float z0 = Xs[o], z1 = Xs[o + 1], z2 = Xs[o + 2];
#pragma unroll
      for (int j = 0; j < 3; ++j) {
        Kb[c * 3 + j] = sIpw[(3 + j) * 3 + 0] * z0 + sIpw[(3 + j) * 3 + 1] * z1 +
                        sIpw[(3 + j) * 3 + 2] * z2 + sIpb[3 + j];
        Vb[c * 3 + j] = sIpw[(6 + j) * 3 + 0] * z0 + sIpw[(6 + j) * 3 + 1] * z1 +
                        sIpw[(6 + j) * 3 + 2] * z2 + sIpb[6 + j];
      }
    }
    __syncthreads();
    int n = base + m;
    if (lane < NMAXQ && n < N) {
      int q = lane;
      int cnt = counts[n];
      float Q0 = 0.f, Q1 = 0.f, Q2 = 0.f;
      if (q < cnt) {
        int fi = frag_idx[n * NMAXQ + q];
        Q0 = x_fine[(long)fi * 3 + 0];
        Q1 = x_fine[(long)fi * 3 + 1];
        Q2 = x_fine[(long)fi * 3 + 2];
      }
      float qv0 = sIpw[0] * Q0 + sIpw[1] * Q1 + sIpw[2] * Q2 + sIpb[0];
      float qv1 = sIpw[3] * Q0 + sIpw[4] * Q1 + sIpw[5] * Q2 + sIpb[1];
      float qv2 = sIpw[6] * Q0 + sIpw[7] * Q1 + sIpw[8] * Q2 + sIpb[2];
      float mx = -1e30f;
      for (int k = 0; k < 64; ++k) {
        float s = (qv0 * Kb[k * 3] + qv1 * Kb[k * 3 + 1] + qv2 * Kb[k * 3 + 2]) * rs3;
        mx = fmaxf(mx, s);
      }
      float se = 0.f, o0 = 0.f, o1 = 0.f, o2 = 0.f;
      for (int k = 0; k < 64; ++k) {
        float s = (qv0 * Kb[k * 3] + qv1 * Kb[k * 3 + 1] + qv2 * Kb[k * 3 + 2]) * rs3;
        float w = __expf(s - mx);
        se += w;
        o0 += w * Vb[k * 3]; o1 += w * Vb[k * 3 + 1]; o2 += w * Vb[k * 3 + 2];
      }
      float inv = 1.f / se;
      o0 *= inv; o1 *= inv; o2 *= inv;
      float r0 = sOw[0] * o0 + sOw[1] * o1 + sOw[2] * o2 + sOb[0];
      float r1 = sOw[3] * o0 + sOw[4] * o1 + sOw[5] * o2 + sOb[1];
      float r2 = sOw[6] * o0 + sOw[7] * o1 + sOw[8] * o2 + sOb[2];
      long off = ((long)n * NMAXQ + q) * 3;
      attn_out[off + 0] = r0; attn_out[off + 1] = r1; attn_out[off + 2] = r2;
    }
    __syncthreads();
  }
}

// ===========================================================================
// Kernel 2: fragment VN block. v-dim is 1 here, so vn_relu's dot/dsq are
// per-channel scalars; W3_2 @ W3_1 folds into a 3x3 per block.
// ===========================================================================
__global__ __launch_bounds__(256)
void frag_vn_kernel(const float* __restrict__ attn_out,
                    const float* __restrict__ W3_1,
                    const float* __restrict__ W3_2,
                    const float* __restrict__ W3_d,
                    const float* __restrict__ g2v,
                    const float* __restrict__ b2v,
                    const int*   __restrict__ gather_b,
                    const int*   __restrict__ gather_p,
                    float*       __restrict__ x_cc,
                    long M)
{
  __shared__ float C33[9], D33[9], sg[3], sb[3];
  int t = threadIdx.x;
  if (t < 9) {
    int o = t / 3, ii = t % 3;
    float s = 0.f;
    for (int c = 0; c < 64; ++c) s += W3_2[o * 64 + c] * W3_1[c * 3 + ii];
    C33[t] = s; D33[t] = W3_d[t];
  }
  if (t < 3) { sg[t] = g2v[t]; sb[t] = b2v[t]; }
  __syncthreads();

  long i = (long)blockIdx.x * 256 + t;
  if (i >= M) return;
  int b = gather_b[i], p = gather_p[i];
  const float* a = attn_out + ((long)b * NMAXQ + p) * 3;
  float x0 = a[0], x1 = a[1], x2 = a[2];
  float h0 = C33[0] * x0 + C33[1] * x1 + C33[2] * x2;
  float h1 = C33[3] * x0 + C33[4] * x1 + C33[5] * x2;
  float h2 = C33[6] * x0 + C33[7] * x1 + C33[8] * x2;
  float d0 = D33[0] * h0 + D33[1] * h1 + D33[2] * h2;
  float d1 = D33[3] * h0 + D33[4] * h1 + D33[5] * h2;
  float d2 = D33[6] * h0 + D33[7] * h1 + D33[8] * h2;
  // per-channel dot (v axis has size 1)
  float t0 = h0 * d0, t1 = h1 * d1, t2 = h2 * d2;
  float q0 = d0 * d0, q1 = d1 * d1, q2 = d2 * d2;
  float y0 = (t0 >= 0.f) ? h0 : h0 - t0 / (q0 + VN_EPS) * d0;
  float y1 = (t1 >= 0.f) ? h1 : h1 - t1 / (q1 + VN_EPS) * d1;
  float y2 = (t2 >= 0.f) ? h2 : h2 - t2 / (q2 + VN_EPS) * d2;
  float r0 = y0 + x0, r1 = y1 + x1, r2 = y2 + x2;
  float n0 = fabsf(r0), n1 = fabsf(r1), n2 = fabsf(r2);
  float mean = (n0 + n1 + n2) * (1.f / 3.f);
  float var = (n0 * n0 + n1 * n1 + n2 * n2) * (1.f / 3.f) - mean * mean;
  float rs = rsqrtf(var + LN_EPS);
  float l0 = (n0 - mean) * rs * sg[0] + sb[0];
  float l1 = (n1 - mean) * rs * sg[1] + sb[1];
  float l2 = (n2 - mean) * rs * sg[2] + sb[2];
  x_cc[i * 3 + 0] = r0 * l0 / (n0 + VN_EPS);
  x_cc[i * 3 + 1] = r1 * l1 / (n1 + VN_EPS);
  x_cc[i * 3 + 2] = r2 * l2 / (n2 + VN_EPS);
}

// ===========================================================================
// Kernel 3: h_cc = ref_feat(M x 64) @ W_emb.T(64 x 64) + b_emb. HBM-bound:
// async global->LDS row staging (ASYNCcnt path), W_emb transposed once to
// LDS, 16x16x4 f32 WMMA; 16 rows x 64 cols per wave, 8 waves/block.
// ===========================================================================
__global__ __launch_bounds__(256)
void emb_gemm_kernel(const float* __restrict__ ref_feat,
                     const float* __restrict__ W_emb,
                     const float* __restrict__ b_emb,
                     float*       __restrict__ h_cc,
                     long M)
{
  __shared__ __attribute__((aligned(16))) float Bt[64 * 64];       // W_emb^T
  __shared__ __attribute__((aligned(16))) float rowbuf[8][16 * 64];
  const int t = threadIdx.x, lane = t & 31, w = t >> 5;
  const int hh = lane >> 4, nn = lane & 15;

  for (int idx = t; idx < 4096; idx += 256) {
    int k = idx >> 6, n = idx & 63;
    Bt[idx] = W_emb[n * 64 + k];          // Bt[k][n] = W_emb.T[k][n]
  }

  long r0 = (long)blockIdx.x * 128 + (long)w * 16;
  const long nrow4 = M * 16;              // total float4s in ref_feat
  {
    const float4* s4 = (const float4*)ref_feat;
    float4* d4 = (float4*)rowbuf[w];
    long base4 = r0 * 16;                 // 16 float4 per 64-float row
#if HAVE_ASYNC_LDS
#pragma unroll
    for (int j = 0; j < 8; ++j) {
      long gi = base4 + lane + j * 32;
      if (gi >= nrow4) gi = 0;            // keep address in-bounds (rows
                                          // beyond M are masked at the store)
      async_b128(s4 + gi, d4 + lane + j * 32);
    }
    async_wait0();
#else
#pragma unroll
    for (int j = 0; j < 8; ++j) {
      long gi = base4 + lane + j * 32;
      float4 v = make_float4(0.f, 0.f, 0.f, 0.f);
      if (gi < nrow4) v = s4[gi];
      d4[lane + j * 32] = v;
    }
#endif
  }
  // speculative prefetch of the next 128-row block (global_prefetch_b8)
  if (r0 + 128 < M)
    __builtin_prefetch(ref_feat + (r0 + 128) * 64 + lane * 16, 0, 0);
  __syncthreads();

  v8f acc0 = {0.f,0.f,0.f,0.f,0.f,0.f,0.f,0.f};
  v8f acc1 = acc0, acc2 = acc0, acc3 = acc0;
  const float* rb = rowbuf[w];
#pragma unroll
  for (int kk = 0; kk < 16; ++kk) {
    int k0 = kk * 4 + 2 * hh;
    v2f a;  a.x  = rb[nn * 64 + k0];       a.y  = rb[nn * 64 + k0 + 1];
    v2f b0; b0.x = Bt[k0 * 64 + nn];       b0.y = Bt[(k0 + 1) * 64 + nn];
    v2f b1; b1.x = Bt[k0 * 64 + 16 + nn];  b1.y = Bt[(k0 + 1) * 64 + 16 + nn];
    v2f b2; b2.x = Bt[k0 * 64 + 32 + nn];  b2.y = Bt[(k0 + 1) * 64 + 32 + nn];
    v2f b3; b3.x = Bt[k0 * 64 + 48 + nn];  b3.y = Bt[(k0 + 1) * 64 + 48 + nn];
    acc0 = wmma_f32_4(a, b0, acc0);
    acc1 = wmma_f32_4(a, b1, acc1);
    acc2 = wmma_f32_4(a, b2, acc2);
    acc3 = wmma_f32_4(a, b3, acc3);
  }
#pragma unroll
  for (int p = 0; p < 8; ++p) {
    long row = r0 + p + 8 * hh;
    if (row < M) {
      h_cc[row * 64 + nn]      = acc0[p] + b_emb[nn];
      h_cc[row * 64 + 16 + nn] = acc1[p] + b_emb[16 + nn];
      h_cc[row * 64 + 32 + nn] = acc2[p] + b_emb[32 + nn];
      h_cc[row * 64 + 48 + nn] = acc3[p] + b_emb[48 + nn];
    }
  }
}

// ===========================================================================
extern "C" void kernel_launch(void* const* d_in, const int* in_sizes, int n_in,
                              void* d_out, int out_size, void* d_ws, size_t ws_size,
                              hipStream_t stream)
{
  const float* Z_V    = (const float*)d_in[0];
  const float* x_fine = (const float*)d_in[1];
  const float* ref    = (const float*)d_in[2];
  const float* W_ff1  = (const float*)d_in[3];
  const float* W_ff2  = (const float*)d_in[4];
  const float* W_ffd  = (const float*)d_in[5];
  const float* g1v    = (const float*)d_in[6];
  const float* b1v    = (const float*)d_in[7];
  const float* ipw    = (const float*)d_in[8];
  const float* ipb    = (const float*)d_in[9];
  const float* outw   = (const float*)d_in[10];
  const float* outb   = (const float*)d_in[11];
  const float* W3_1   = (const float*)d_in[12];
  const float* W3_2   = (const float*)d_in[13];
  const float* W3_d   = (const float*)d_in[14];
  const float* g2v    = (const float*)d_in[15];
  const float* b2v    = (const float*)d_in[16];
  const float* W_emb  = (const float*)d_in[17];
  const float* b_emb  = (const float*)d_in[18];
  const int* frag_idx = (const int*)d_in[19];
  const int* counts   = (const int*)d_in[20];
  const int* gather_b = (const int*)d_in[21];
  const int* gather_p = (const int*)d_in[22];

  const int  N = in_sizes[0] / (FCH * 3);
  const long M = (long)in_sizes[1] / 3;

  float* attn_ws = (float*)d_ws;                 // N*16*3 floats scratch
  float* x_cc = (float*)d_out;                   // (M,3)
  float* h_cc = (float*)d_out + M * 3;           // (M,64)

  const int blocks1 = (N + MOLS_PER_WAVE - 1) / MOLS_PER_WAVE;
  vn_attn_kernel<<<blocks1, 32, 0, stream>>>(
      Z_V, x_fine, W_ff1, W_ff2, W_ffd, g1v, b1v, ipw, ipb, outw, outb,
      frag_idx, counts, attn_ws, N);

  emb_gemm_kernel<<<(int)((M + 127) / 128), 256, 0, stream>>>(
      ref, W_emb, b_emb, h_cc, M);

  frag_vn_kernel<<<(int)((M + 255) / 256), 256, 0, stream>>>(
      attn_ws, W3_1, W3_2, W3_d, g2v, b2v, gather_b, gather_p, x_cc, M);
}